// Probabilistic_DAG_Generator_From_Roots_35046933135804
// MI455X (gfx1250) — compile-verified
//
#include <hip/hip_runtime.h>
#include <stdint.h>

// N = 2048 nodes; bit rows are 64 x u32 words.
#define NN   2048
#define NWRD 64

typedef unsigned int u32x4 __attribute__((ext_vector_type(4)));
typedef int          i32x8 __attribute__((ext_vector_type(8)));
typedef int          i32x4 __attribute__((ext_vector_type(4)));

#if defined(__AMDGCN__) && __has_builtin(__builtin_amdgcn_tensor_load_to_lds) && __has_builtin(__builtin_amdgcn_s_wait_tensorcnt)
#define HAVE_TDM 1
#else
#define HAVE_TDM 0
#if defined(__AMDGCN__)
#warning "CDNA5 TDM builtins unavailable on this toolchain; using plain LDS loads"
#endif
#endif

// jax.nn.log_sigmoid(x) = -softplus(-x) = min(x,0) - log1p(exp(-|x|))
__device__ __forceinline__ float log_sigmoid_f(float x) {
  return fminf(x, 0.0f) - log1pf(expf(-fabsf(x)));
}

__device__ __forceinline__ uint32_t ballot32(bool p) {
#if defined(__AMDGCN__)
  return __builtin_amdgcn_ballot_w32(p);   // wave32 ballot (gfx1250)
#else
  (void)p; return 0u;                       // host pass stub
#endif
}

#if HAVE_TDM
// TDM: async DMA of one 64-word (256B) row from global memory into LDS.
// D# per CDNA5 ISA ch.8: group0 = {count, lds_addr, gaddr_lo, gaddr_hi|type=2},
// group1 = data_size=4B, tensor_dim0=64, tensor_dim1=1, tile_dim0=64, tile_dim1=1,
// tensor_dim0_stride=64. Groups 2/3 zero (<=2D tensor). Completion via TENSORcnt.
__device__ __forceinline__ void tdm_issue_row64(uint32_t lds_addr, const uint32_t* gsrc) {
  uint64_t ga = (uint64_t)(uintptr_t)gsrc;
  u32x4 g0;
  g0.x = 1u;                                                 // count=1, user descriptor
  g0.y = lds_addr;                                           // lds_addr -> bits [63:32]
  g0.z = (unsigned)(ga & 0xFFFFFFFFu);                       // global_addr [95:64]
  g0.w = (unsigned)((ga >> 32) & 0x01FFFFFFu) | (2u << 30);  // addr[56:32] | type=2
  i32x8 g1;
  g1[0] = 0x00020000;        // workgroup_mask=0, data_size=2 (4-byte elements)
  g1[1] = (int)(64u << 16);  // tensor_dim0 = 64  (bits 63:48)
  g1[2] = (int)(1u << 16);   // tensor_dim0_hi=0, tensor_dim1 = 1 (bits 95:80)
  g1[3] = (int)(64u << 16);  // tensor_dim1_hi=0, tile_dim0 = 64 (bits 127:112)
  g1[4] = 1;                 // tile_dim1 = 1, tile_dim2 = 0
  g1[5] = 64;                // tensor_dim0_stride = 64
  g1[6] = 0;
  g1[7] = 0;
  i32x4 gz = {0, 0, 0, 0};
#if __clang_major__ >= 23
  i32x8 gz8 = {0, 0, 0, 0, 0, 0, 0, 0};
  __builtin_amdgcn_tensor_load_to_lds(g0, g1, gz, gz, gz8, 0);
#else
  __builtin_amdgcn_tensor_load_to_lds(g0, g1, gz, gz, 0);
#endif
}
#endif

// ---------------------------------------------------------------------------
// Kernel 1: grid-wide precompute (bandwidth phase: ~48MB in / 17MB out, ~3us
// at 23.3 TB/s -- negligible next to the sequential BFS).
//  - hardE[i][j] bit  = (ls(e)+g0 > ls(-e)+g1), bit-packed via wave32 ballot
//  - anc init         = identity (anc[i][i] = 1)
//  - rootBits[i] bit  = (ls(r)+g0 > ls(-r)+g1)
//  - dag zeroed (harness poisons d_out)
// ---------------------------------------------------------------------------
__global__ void dag_prep_kernel(const float* __restrict__ rl, const float* __restrict__ el,
                                const float* __restrict__ gr, const float* __restrict__ ge,
                                uint32_t* __restrict__ rootBits, uint32_t* __restrict__ hardE,
                                uint32_t* __restrict__ anc, float* __restrict__ dag) {
  int idx = blockIdx.x * blockDim.x + threadIdx.x;   // 0 .. NN*NN-1
  int i = idx >> 11;
  int j = idx & (NN - 1);
  int lane = threadIdx.x & 31;

  float x  = el[idx];
  float y0 = log_sigmoid_f(x)  + ge[(size_t)i * 2 * NN + j];
  float y1 = log_sigmoid_f(-x) + ge[(size_t)i * 2 * NN + NN + j];
  uint32_t w = ballot32(y0 > y1);
  if (lane == 0) {
    int gw = idx >> 5;
    hardE[gw] = w;
    int colw = j >> 5;
    anc[gw] = (colw == (i >> 5)) ? (1u << (i & 31)) : 0u;   // identity row
  }
  dag[idx] = 0.0f;

  if (idx < NN) {                                    // wave-uniform (NN % 32 == 0)
    float r  = rl[idx];
    float z0 = log_sigmoid_f(r)  + gr[idx];
    float z1 = log_sigmoid_f(-r) + gr[NN + idx];
    uint32_t rw = ballot32(z0 > z1);
    if (lane == 0) rootBits[idx >> 5] = rw;
  }
}

// ---------------------------------------------------------------------------
// Kernel 2: sequential BFS, one 1024-thread workgroup (32 waves), 3 barriers
// per pop. Per pop i:
//   children = hardE[i] & ~anc[i] & ~roots     (== dag row, written as 0/1 f32)
//   anc[j]  |= anc[i]  for each child j        (bit-packed, 16 rows x 64 words)
//   enqueue new children in ascending index order (FIFO semantics).
// The next pop's hardE row is TDM-DMA'd into a double-buffered LDS slot during
// the ancestor-OR phase and only waited on (TENSORcnt) at the next loop top,
// so the DMA latency is fully hidden.
// ---------------------------------------------------------------------------
__global__ void __launch_bounds__(1024)
dag_bfs_kernel(const uint32_t* __restrict__ rootBits, const uint32_t* __restrict__ hardE,
               uint32_t* __restrict__ anc, float* __restrict__ dag) {
  __shared__ uint32_t rootW[NWRD];
  __shared__ uint32_t enqW[NWRD];
  __shared__ uint32_t ancI[NWRD];
  __shared__ uint32_t childW[NWRD];
#if HAVE_TDM
  __shared__ uint32_t hardW[2][NWRD];    // double-buffered TDM landing zone
#endif
  __shared__ int queue[NN];
  __shared__ int childList[NN];
  __shared__ int sHead, sTail;
  __shared__ int sCC[2];                 // child counter, ping-ponged by parity

  int tid = threadIdx.x;

  if (tid < NWRD) { rootW[tid] = rootBits[tid]; enqW[tid] = rootBits[tid]; }
  if (tid == 0) {
    int t = 0;
    for (int w = 0; w < NWRD; ++w) {
      uint32_t m = rootBits[w];
      while (m) { int b = __ffs(m) - 1; queue[t++] = w * 32 + b; m &= m - 1; }
    }
    sTail = t; sHead = 0; sCC[0] = 0; sCC[1] = 0;
#if HAVE_TDM
    if (t > 0)   // prime the pipeline: DMA first pop's hardE row into buffer 0
      tdm_issue_row64((uint32_t)(uintptr_t)&hardW[0][0], &hardE[(size_t)queue[0] * NWRD]);
#endif
  }
  __syncthreads();

  for (int it = 0; it < NN; ++it) {
    if (sHead >= sTail) break;           // uniform: no further state change possible
    int i = queue[sHead];
    int cur = it & 1;

#if HAVE_TDM
    if (tid == 0) __builtin_amdgcn_s_wait_tensorcnt(0);  // wave0 drains the DMA
    __syncthreads();                     // sync A: hardW[cur] visible to all waves
#endif

    // ---- 64 word-owner threads: childW + dag row + child list (same-thread data)
    if (tid < NWRD) {
      uint32_t a = anc[(size_t)i * NWRD + tid];   // coherent WGP-scope load (anc is hot RW)
      ancI[tid] = a;
#if HAVE_TDM
      uint32_t c = hardW[cur][tid] & ~a & ~rootW[tid];
#else
      uint32_t c = hardE[(size_t)i * NWRD + tid] & ~a & ~rootW[tid];
#endif
      childW[tid] = c;
      float4* drow = (float4*)(dag + (size_t)i * NN + tid * 32);   // 128B-aligned
      #pragma unroll
      for (int b = 0; b < 32; b += 4) {
        float4 v;
        v.x = (float)((c >> (b + 0)) & 1u);
        v.y = (float)((c >> (b + 1)) & 1u);
        v.z = (float)((c >> (b + 2)) & 1u);
        v.w = (float)((c >> (b + 3)) & 1u);
        drow[b >> 2] = v;
      }
      uint32_t m = c;                    // unordered compact list (order-free OR phase)
      while (m) {
        int b = __ffs(m) - 1;
        int slot = atomicAdd(&sCC[cur], 1);
        childList[slot] = tid * 32 + b;
        m &= m - 1;
      }
    }
    __syncthreads();                     // sync B: childW / childList complete

    int nCh = sCC[cur];

    // ---- tid0: ordered FIFO append, head++, prefetch + TDM-issue for next pop.
    //      Runs concurrently with the 1024-thread ancestor fan-out below.
    if (tid == 0) {
      int t = sTail;
      for (int w = 0; w < NWRD; ++w) {
        uint32_t m = childW[w] & ~enqW[w];
        enqW[w] |= m;
        while (m) { int b = __ffs(m) - 1; queue[t++] = w * 32 + b; m &= m - 1; }
      }
      sTail = t;
      int h = sHead + 1;
      sHead = h;
      sCC[cur ^ 1] = 0;                  // reset the *other* parity's counter
      if (h < t) {
        int nx = queue[h];
        __builtin_prefetch(&anc[(size_t)nx * NWRD]);   // global_prefetch_b8
#if HAVE_TDM
        // async DMA of next pop's hardE row; waited at next loop top (TENSORcnt).
        tdm_issue_row64((uint32_t)(uintptr_t)&hardW[cur ^ 1][0],
                        &hardE[(size_t)nx * NWRD]);
#else
        __builtin_prefetch(&hardE[(size_t)nx * NWRD]);
#endif
      }
    }

    // ---- ancestor fan-out: anc[j] |= anc[i], 16 child rows x 64 words in flight
    int rowSlot = tid >> 6;              // 0..15
    int w = tid & 63;
    for (int k = rowSlot; k < nCh; k += 16) {
      int j = childList[k];
      anc[(size_t)j * NWRD + w] |= ancI[w];   // each (j,w) touched by exactly one thread
    }
    __syncthreads();                     // sync C
  }
}

// ---------------------------------------------------------------------------
// Host launcher. Workspace layout (all u32, ~1.0 MB; fully re-initialized by
// dag_prep_kernel every call -> deterministic, graph-replay safe):
//   [0)          hardE    : NN*NWRD words (512 KB)
//   [NN*NWRD)    anc      : NN*NWRD words (512 KB)
//   [2*NN*NWRD)  rootBits : NWRD words (256 B)
// ---------------------------------------------------------------------------
extern "C" void kernel_launch(void* const* d_in, const int* in_sizes, int n_in,
                              void* d_out, int out_size, void* d_ws, size_t ws_size,
                              hipStream_t stream) {
  (void)in_sizes; (void)n_in; (void)out_size; (void)ws_size;
  const float* rl = (const float*)d_in[0];   // root_logits  [N]
  const float* el = (const float*)d_in[1];   // edge_logits  [N,N]
  const float* gr = (const float*)d_in[2];   // g_root       [2,N]
  const float* ge = (const float*)d_in[3];   // g_edge       [N,2,N]
  float* dag = (float*)d_out;                // [N,N] f32

  uint32_t* hardE    = (uint32_t*)d_ws;
  uint32_t* anc      = hardE + (size_t)NN * NWRD;
  uint32_t* rootBits = anc + (size_t)NN * NWRD;

  dag_prep_kernel<<<(NN * NN) / 256, 256, 0, stream>>>(rl, el, gr, ge,
                                                       rootBits, hardE, anc, dag);
  dag_bfs_kernel<<<1, 1024, 0, stream>>>(rootBits, hardE, anc, dag);
}